// YOLOLayer_44925357916399
// MI455X (gfx1250) — compile-verified
//
#include <hip/hip_runtime.h>
#include <cstdint>

// YOLO decode for MI455X (gfx1250): pure HBM-streaming problem (~176MB @ 23.3TB/s).
// Strategy: async global->LDS DMA (CDNA5 GLOBAL_LOAD_ASYNC_TO_LDS_B128) to stage the
// channel-major tile, then transform + coalesced non-temporal channel-minor stores.

namespace {
constexpr int kG      = 52;            // grid size
constexpr int kGG     = kG * kG;       // 2704 positions
constexpr int kA      = 3;             // anchors
constexpr int kCH     = 85;            // 5 + 80 classes
constexpr int kB      = 32;            // batch
constexpr int kTP     = 128;           // positions per tile
constexpr int kTPP    = 132;           // LDS row pitch (mult of 4 -> 16B align; 2-way banks)
constexpr int kBLOCK  = 256;           // threads per block (8 waves)
constexpr int kNTILES = (kGG + kTP - 1) / kTP;  // 22
constexpr float kStride = 8.0f;        // 416 / 52
constexpr float kLog2e  = 1.4426950408889634f;
}

__global__ __launch_bounds__(kBLOCK) void yolo_decode_kernel(
    const float* __restrict__ x, float* __restrict__ out) {
  __shared__ __align__(16) float raw[kCH * kTPP];  // 44,880 B

  const int bid  = blockIdx.x;
  const int tile = bid % kNTILES;
  const int ba   = bid / kNTILES;          // b*3 + a
  const int a    = ba % kA;
  const int pos0 = tile * kTP;
  const int V    = (kGG - pos0 < kTP) ? (kGG - pos0) : kTP;  // 128 or 16 (mult of 4)
  const int tid  = threadIdx.x;

  const float aw = (a == 0) ? 10.0f : (a == 1) ? 16.0f : 33.0f;
  const float ah = (a == 0) ? 13.0f : (a == 1) ? 30.0f : 23.0f;

  // channel rows for this (b,a): b*255 + a*85 == 85*ba
  const float* src = x + (size_t)ba * kCH * kGG + pos0;
  __builtin_prefetch(src, 0, 0);  // global_prefetch_b8 (gfx1250)

  // ---- Stage 1: async DMA tile into LDS, layout raw[ch*kTPP + pos] ----
  constexpr int kChunks = kCH * (kTP / 4);  // 2720 x 16B
  for (int i = tid; i < kChunks; i += kBLOCK) {
    int ch = i >> 5;               // i / 32
    int p  = (i & 31) << 2;        // float offset within row, 16B-aligned
    if (p + 4 > V) p = V - 4;      // tail clamp, stays 16B-aligned (V mult of 4)
    uint32_t lds_addr = (uint32_t)(uintptr_t)(&raw[ch * kTPP + p]);
    const float* g = src + (size_t)ch * kGG + p;
    asm volatile("global_load_async_to_lds_b128 %0, %1, off"
                 :: "v"(lds_addr), "v"(g)
                 : "memory");
  }
  asm volatile("s_wait_asynccnt 0" ::: "memory");
  __syncthreads();

  // ---- Stage 2: transform + coalesced NT stores (out[b][a*GG+pos][ch]) ----
  float* dst = out + ((size_t)ba * kGG + pos0) * kCH;
  const int nElem = V * kCH;
  for (int e = tid; e < nElem; e += kBLOCK) {
    int pos = e / kCH;             // magic-mul division
    int ch  = e - pos * kCH;
    float v = raw[ch * kTPP + pos];
    // sigmoid via raw v_exp_f32 / v_rcp_f32 (4 VALU ops)
    float sig = __builtin_amdgcn_rcpf(1.0f + __builtin_amdgcn_exp2f(v * -kLog2e));
    float val;
    if (ch >= 4) {
      val = sig;                                   // conf + classes
    } else if (ch == 0) {
      int gpos = pos0 + pos;
      int gx = gpos % kG;
      val = (sig + (float)gx) * kStride;           // x
    } else if (ch == 1) {
      int gpos = pos0 + pos;
      int gy = gpos / kG;
      val = (sig + (float)gy) * kStride;           // y
    } else {
      float ex = __builtin_amdgcn_exp2f(v * kLog2e);
      val = ex * ((ch == 2) ? aw : ah);            // w / h (anchor/stride*stride = anchor)
    }
    __builtin_nontemporal_store(val, &dst[e]);     // global_store_b32 th:TH_STORE_NT
  }
}

__global__ void fill_zero_kernel(float* p, long long start, long long n) {
  long long i = (long long)blockIdx.x * blockDim.x + threadIdx.x;
  if (i < n) __builtin_nontemporal_store(0.0f, &p[start + i]);
}

extern "C" void kernel_launch(void* const* d_in, const int* in_sizes, int n_in,
                              void* d_out, int out_size, void* d_ws, size_t ws_size,
                              hipStream_t stream) {
  (void)in_sizes; (void)n_in; (void)d_ws; (void)ws_size;
  const float* x = (const float*)d_in[0];
  float* out = (float*)d_out;

  const long long N = (long long)kB * kA * kGG * kCH;  // 22,064,640
  dim3 grid(kB * kA * kNTILES);                        // 2112 blocks
  yolo_decode_kernel<<<grid, kBLOCK, 0, stream>>>(x, out);

  if ((long long)out_size > N) {                       // trailing scalar 0 of the tuple
    long long n = (long long)out_size - N;
    int blocks = (int)((n + 255) / 256);
    fill_zero_kernel<<<blocks, 256, 0, stream>>>(out, N, n);
  }
}